// BoundingBoxExtractor_42494406427148
// MI455X (gfx1250) — compile-verified
//
#include <hip/hip_runtime.h>
#include <hip/hip_bf16.h>
#include <stdint.h>

// Problem constants (from reference): B,H,W = 64,512,512 ; SCALE = 1
#define B_    64
#define H_    512
#define W_    512
#define TPB   256                           // 8 wave32 waves
#define SPLIT 16                            // blocks per batch
#define ROWS_PER_BLK (H_ / SPLIT)           // 32 rows
#define F4_PER_BLK   (ROWS_PER_BLK * (W_/4))// 4096 float4 per block
#define NITER        (F4_PER_BLK / TPB)     // 16 iterations per thread
#define STAGES 8                            // LDS ring depth (32 KB) -> 7 loads in flight
#define DEPTH  (STAGES - 1)                 // outstanding async loads in steady state
#define BIGI   0x3FFFFFFF

// ---- gfx1250 async global->LDS path (inline asm, ASYNCcnt-tracked) --------
__device__ __forceinline__ void async_ld_b128(uint32_t lds_addr, const float4* g) {
#if defined(__gfx1250__)
    // VDST = VGPR holding LDS byte address, VADDR = 64-bit global address, no SADDR
    asm volatile("global_load_async_to_lds_b128 %0, %1, off"
                 :: "v"(lds_addr), "v"(g)
                 : "memory");
#else
    (void)lds_addr; (void)g;
#endif
}

#if defined(__gfx1250__)
#define WAIT_ASYNC(n) asm volatile("s_wait_asynccnt " #n ::: "memory")
#else
#define WAIT_ASYNC(n)
#endif

// Async loads complete in order: allowing `rem` outstanding guarantees the
// oldest (current stage's) load has landed in LDS.
__device__ __forceinline__ void wait_async_upto(int rem) {
    if      (rem >= 7) { WAIT_ASYNC(7); }
    else if (rem == 6) { WAIT_ASYNC(6); }
    else if (rem == 5) { WAIT_ASYNC(5); }
    else if (rem == 4) { WAIT_ASYNC(4); }
    else if (rem == 3) { WAIT_ASYNC(3); }
    else if (rem == 2) { WAIT_ASYNC(2); }
    else if (rem == 1) { WAIT_ASYNC(1); }
    else               { WAIT_ASYNC(0); }
}

// ---- workspace init: ws[b*5 + {0..4}] = {rmin, rmax, cmin, cmax, nonedge} -
__global__ void bbox_init(int* __restrict__ ws) {
    int i = blockIdx.x * blockDim.x + threadIdx.x;
    if (i < B_ * 5) {
        int k = i % 5;
        ws[i] = (k == 0 || k == 2) ? BIGI : ((k == 1 || k == 3) ? -1 : 0);
    }
}

// ---- main streaming reduction --------------------------------------------
__global__ __launch_bounds__(TPB) void bbox_scan(const float* __restrict__ mask,
                                                 int* __restrict__ ws) {
    __shared__ float4 stage[STAGES * TPB];  // 32 KB ring buffer
    __shared__ int    s_red[5];

    const int t   = threadIdx.x;
    const int blk = blockIdx.x;
    const int b   = blk >> 4;               // / SPLIT
    const int s   = blk & (SPLIT - 1);

    if (t == 0) {
        s_red[0] = BIGI; s_red[1] = -1; s_red[2] = BIGI; s_red[3] = -1; s_red[4] = 0;
    }
    __syncthreads();

    const float4* gbase = (const float4*)(mask
                        + (size_t)b * ((size_t)H_ * W_)
                        + (size_t)s * ROWS_PER_BLK * W_);

    // LDS byte address of this thread's slot in stage 0 (low 32 bits of the
    // flat shared pointer == LDS offset; SHARED aperture occupies [63:32]).
    uint32_t lds0 = (uint32_t)(uintptr_t)(&stage[t]);

    // Prologue: DEPTH = 7 stages in flight (112 B/thread outstanding)
    #pragma unroll
    for (int p = 0; p < DEPTH; ++p)
        async_ld_b128(lds0 + (uint32_t)(p * TPB * 16), gbase + p * TPB + t);

    int lrmin = BIGI, lrmax = -1, lcmin = BIGI, lcmax = -1, nonedge = 0;

    #pragma unroll
    for (int i = 0; i < NITER; ++i) {
        if (i + DEPTH < NITER)
            async_ld_b128(lds0 + (uint32_t)(((i + DEPTH) & (STAGES - 1)) * TPB * 16),
                          gbase + (i + DEPTH) * TPB + t);

        wait_async_upto(NITER - 1 - i);     // folds to a constant per unrolled iter

        float4 v = stage[(i & (STAGES - 1)) * TPB + t];   // ds_load_b128, own slot

        const int f = i * TPB + t;                        // float4 index in block tile
        const int R = s * ROWS_PER_BLK + (f >> 7);        // W/4 = 128 float4 per row
        const int C = (f & 127) << 2;

        int e0 = v.x >= 0.5f;
        int e1 = v.y >= 0.5f;
        int e2 = v.z >= 0.5f;
        int e3 = v.w >= 0.5f;
        if (e0 | e1 | e2 | e3) {
            lrmin = min(lrmin, R);
            lrmax = max(lrmax, R);
            int clo = e0 ? C     : (e1 ? C + 1 : (e2 ? C + 2 : C + 3));
            int chi = e3 ? C + 3 : (e2 ? C + 2 : (e1 ? C + 1 : C));
            lcmin = min(lcmin, clo);
            lcmax = max(lcmax, chi);
        }
        if (!(e0 & e1 & e2 & e3)) nonedge = 1;
    }

    // Block combine in LDS, then one set of global atomics per block.
    atomicMin(&s_red[0], lrmin);
    atomicMax(&s_red[1], lrmax);
    atomicMin(&s_red[2], lcmin);
    atomicMax(&s_red[3], lcmax);
    atomicOr (&s_red[4], nonedge);
    __syncthreads();

    if (t == 0) {
        int* wb = ws + b * 5;
        atomicMin(&wb[0], s_red[0]);
        atomicMax(&wb[1], s_red[1]);
        atomicMin(&wb[2], s_red[2]);
        atomicMax(&wb[3], s_red[3]);
        atomicOr (&wb[4], s_red[4]);
    }
}

// ---- finalize: apply the reference's -1 fixups, write both outputs --------
__global__ void bbox_final(const int* __restrict__ ws, float* __restrict__ out) {
    int b = blockIdx.x * blockDim.x + threadIdx.x;
    if (b >= B_) return;

    int rmin = ws[b * 5 + 0], rmax = ws[b * 5 + 1];
    int cmin = ws[b * 5 + 2], cmax = ws[b * 5 + 3];
    int ne   = ws[b * 5 + 4];

    bool has   = (rmax >= 0);                 // any edge pixel at all
    float y_max = has ? (float)rmax : 0.0f;   // x_max += (x_max == -1)
    float x_max = has ? (float)cmax : 0.0f;
    float rminf = has ? (float)rmin : 3.0e38f;
    float cminf = has ? (float)cmin : 3.0e38f;

    float y_min, x_min;
    if (ne) { // non-edge pixels contribute (y_max - 1) / (x_max - 1) to the min
        y_min = fminf(rminf, y_max - 1.0f);
        x_min = fminf(cminf, x_max - 1.0f);
    } else {  // every pixel is an edge
        y_min = rminf;
        x_min = cminf;
    }

    // Output 0: bounding_boxes_scaled (B,4) ; Output 1: * SCALE (== 1)
    out[b * 4 + 0] = x_min;
    out[b * 4 + 1] = y_min;
    out[b * 4 + 2] = x_max;
    out[b * 4 + 3] = y_max;
    const float SC = 1.0f;
    out[B_ * 4 + b * 4 + 0] = x_min * SC;
    out[B_ * 4 + b * 4 + 1] = y_min * SC;
    out[B_ * 4 + b * 4 + 2] = x_max * SC;
    out[B_ * 4 + b * 4 + 3] = y_max * SC;
}

extern "C" void kernel_launch(void* const* d_in, const int* in_sizes, int n_in,
                              void* d_out, int out_size, void* d_ws, size_t ws_size,
                              hipStream_t stream) {
    const float* mask = (const float*)d_in[0];  // (64,1,512,512) f32
    int*   ws  = (int*)d_ws;                    // 64*5 ints
    float* out = (float*)d_out;                 // 2 * (64,4) f32

    bbox_init <<<1, 512, 0, stream>>>(ws);
    bbox_scan <<<B_ * SPLIT, TPB, 0, stream>>>(mask, ws);
    bbox_final<<<1, 64, 0, stream>>>(ws, out);
}